// ES_Normalize_309237645779
// MI455X (gfx1250) — compile-verified
//
#include <hip/hip_runtime.h>
#include <hip/hip_bf16.h>

typedef __attribute__((ext_vector_type(2))) float v2f;
typedef __attribute__((ext_vector_type(8))) float v8f;

// Problem constants (from reference setup_inputs / ALPHA)
#define BATCH 64
#define SEQL  2048
#define DIM   512
#define ALPHA_C 0.3f

// Segmented scan: 4 segments of 512 steps, 64-step warm-up halo.
// 0.3^64 ~ 3e-34 << fp32 relative eps, so halo warm-up reproduces the
// exact-scan values to full fp32 precision for same-scale data.
#define SEG   512
#define HALO  64
#define NSEG  (SEQL / SEG)
#define WAVES_PER_BLOCK 8

__global__ __launch_bounds__(256) void es_scan_wmma(const float* __restrict__ x,
                                                    float* __restrict__ y,
                                                    float* __restrict__ ylast) {
    const int lane = threadIdx.x & 31;
    const int wave = blockIdx.x * WAVES_PER_BLOCK + (threadIdx.x >> 5);

    // wave -> (b, d-tile, segment)
    const int seg = wave & (NSEG - 1);          // 0..3
    const int dt  = (wave >> 2) & 31;           // 0..31 (DIM/16 tiles)
    const int b   = wave >> 7;                  // 0..63
    const int d0  = dt * 16;

    const int hi = lane >> 4;                   // lane half: 0 or 1
    const int n  = lane & 15;                   // column (channel) / row-in-lane
    const int m  = n;                           // A-matrix row index for this lane

    // a^e via exponent bits (pure selects, stays in VGPRs)
    const float a1 = ALPHA_C, a2 = a1 * a1, a4 = a2 * a2, a8 = a4 * a4, a16 = a8 * a8;
    auto apow = [&](int e) -> float {
        float r = 1.0f;
        if (e & 1)  r *= a1;
        if (e & 2)  r *= a2;
        if (e & 4)  r *= a4;
        if (e & 8)  r *= a8;
        if (e & 16) r *= a16;
        return r;
    };

    // A operand: T[m][k] = (1-a)*a^(m-k) for k<=m, else 0.
    // f32 16x4 A layout: lane holds row m, VGPR r holds K = 4*kk + r + 2*hi.
    v2f Amat[4];
#pragma unroll
    for (int kk = 0; kk < 4; ++kk) {
        v2f t;
#pragma unroll
        for (int r = 0; r < 2; ++r) {
            const int k = kk * 4 + r + 2 * hi;
            t[r] = (k <= m) ? (1.0f - ALPHA_C) * apow(m - k) : 0.0f;
        }
        Amat[kk] = t;
    }

    // Carry weights: row m_r = r + 8*hi gets a^(m_r+1)
    float pw[8];
#pragma unroll
    for (int r = 0; r < 8; ++r) pw[r] = apow(r + 8 * hi + 1);

    const float* xb = x + ((size_t)b * SEQL) * DIM + d0;
    float*       yb = y + ((size_t)b * SEQL) * DIM + d0;

    const int l_begin = seg * SEG;
    const int start_l = (seg == 0) ? 0 : (l_begin - HALO);
    const int nwarm   = (seg == 0) ? 0 : (HALO / 16);
    const int nblk    = nwarm + SEG / 16;

    // Warm-up carry init: y[start] ~= x[start] (a*x + (1-a)*x); halo absorbs it.
    float carry = xb[(size_t)start_l * DIM + n];

    const int loadoff = (2 * hi) * DIM + n;     // B-operand per-lane offset
    const int outoff  = (8 * hi) * DIM + n;     // C/D per-lane offset

    // B operand (4x16 f32): lane holds column n, VGPR r holds K = 4*kk + r + 2*hi
    auto loadB = [&](v2f* dst, int l0) {
#pragma unroll
        for (int kk = 0; kk < 4; ++kk) {
            const size_t base = (size_t)(l0 + kk * 4) * DIM + loadoff;
            v2f t;
            t[0] = xb[base];
            t[1] = xb[base + DIM];
            dst[kk] = t;
        }
    };

    v2f cur[4], nxt[4] = {};
    loadB(cur, start_l);

    for (int blk = 0; blk < nblk; ++blk) {
        const int l0 = start_l + blk * 16;
        if (blk + 1 < nblk) loadB(nxt, l0 + 16);   // pipeline next tile's loads

        v8f c = {0.f, 0.f, 0.f, 0.f, 0.f, 0.f, 0.f, 0.f};
        // Y_blk = T * X_blk : four chained f32 WMMAs over K = 0..15
        c = __builtin_amdgcn_wmma_f32_16x16x4_f32(false, Amat[0], false, cur[0],
                                                  (short)0, c, false, false);
        c = __builtin_amdgcn_wmma_f32_16x16x4_f32(false, Amat[1], false, cur[1],
                                                  (short)0, c, false, false);
        c = __builtin_amdgcn_wmma_f32_16x16x4_f32(false, Amat[2], false, cur[2],
                                                  (short)0, c, false, false);
        c = __builtin_amdgcn_wmma_f32_16x16x4_f32(false, Amat[3], false, cur[3],
                                                  (short)0, c, false, false);

        // Add decayed carry:  y[m] += a^(m+1) * carry
#pragma unroll
        for (int r = 0; r < 8; ++r) c[r] += pw[r] * carry;

        // New carry = y[row 15] (VGPR 7, lanes 16..31), broadcast to all lanes
        carry = __shfl(c[7], 16 + n, 32);

        if (blk >= nwarm) {
#pragma unroll
            for (int r = 0; r < 8; ++r)
                yb[(size_t)(l0 + r) * DIM + outoff] = c[r];
        }

#pragma unroll
        for (int kk = 0; kk < 4; ++kk) cur[kk] = nxt[kk];
    }

    // Second output: y[:, L-1, :]
    if (seg == NSEG - 1 && lane < 16)
        ylast[(size_t)b * DIM + d0 + n] = carry;
}

extern "C" void kernel_launch(void* const* d_in, const int* in_sizes, int n_in,
                              void* d_out, int out_size, void* d_ws, size_t ws_size,
                              hipStream_t stream) {
    const float* x = (const float*)d_in[0];
    float* y     = (float*)d_out;
    float* ylast = y + (size_t)BATCH * SEQL * DIM;

    const int total_waves = BATCH * (DIM / 16) * NSEG;           // 8192
    const int blocks = total_waves / WAVES_PER_BLOCK;            // 1024
    es_scan_wmma<<<blocks, 256, 0, stream>>>(x, y, ylast);
}